// ProgressiveGaussian2D_24257975288069
// MI455X (gfx1250) — compile-verified
//
#include <hip/hip_runtime.h>

// ---------------------------------------------------------------------------
// 2D Gaussian splatting (forward + screen-space derivative images) on gfx1250.
//
// Strategy:
//   Pre-kernel: fold cholesky -> (a,b,c) conic coefficients and sigmoid(rgb)
//               into an 8-float param block per Gaussian in d_ws (32 KB).
//   Main kernel: one wave32 owns a 16-pixel tile. Loop over 1024 Gaussians;
//               per Gaussian issue ONE V_WMMA_F32_16X16X4_F32:
//                 A (16x4)  : rows = pixels, K = {wgt, -gA*wgt, -gB*wgt,
//                                                (gA*gB - b)*wgt}
//                 B (4x16)  : per-Gaussian routing matrix
//                             row0 -> cols 0..3  = [r,g,b,1]   (out + wsum)
//                             row1 -> cols 4..6  = [r,g,b]     (dx)
//                             row2 -> cols 7..9  = [r,g,b]     (dy)
//                             row3 -> cols 10..12= [r,g,b]     (dxy)
//                 D (16x16) : rows = pixels, cols = output channels 0..12
//               d_out layout is [out(3P) | wsum(P) | dx(3P) | dy(3P) | dxy(3P)]
//               => channel c plane starts at c*P exactly.
// ---------------------------------------------------------------------------

typedef __attribute__((ext_vector_type(2))) float v2f;
typedef __attribute__((ext_vector_type(8))) float v8f;

#define NG   1024
#define IMH  256
#define IMW  256
#define NPIX (IMH * IMW)

__global__ void gauss2d_precompute(const float* __restrict__ means,
                                   const float* __restrict__ chol,
                                   const float* __restrict__ rgb_logits,
                                   float* __restrict__ params) {
  int n = blockIdx.x * blockDim.x + threadIdx.x;
  if (n >= NG) return;
  float l1 = chol[n * 3 + 0] + 0.5f;
  float l2 = chol[n * 3 + 1];
  float l3 = chol[n * 3 + 2] + 0.5f;
  float Sxx = l1 * l1;
  float Sxy = l1 * l2;
  float Syy = l2 * l2 + l3 * l3;
  float d   = l1 * l3;
  float inv = 1.0f / (d * d);
  float a = Syy * inv;
  float b = -Sxy * inv;
  float c = Sxx * inv;
  float r  = 1.0f / (1.0f + __expf(-rgb_logits[n * 3 + 0]));
  float g  = 1.0f / (1.0f + __expf(-rgb_logits[n * 3 + 1]));
  float bl = 1.0f / (1.0f + __expf(-rgb_logits[n * 3 + 2]));
  params[n * 8 + 0] = means[n * 2 + 0];
  params[n * 8 + 1] = means[n * 2 + 1];
  params[n * 8 + 2] = a;
  params[n * 8 + 3] = b;
  params[n * 8 + 4] = c;
  params[n * 8 + 5] = r;
  params[n * 8 + 6] = g;
  params[n * 8 + 7] = bl;
}

__global__ void __launch_bounds__(256)
gauss2d_splat_wmma(const float* __restrict__ params,
                   float* __restrict__ out) {
  const int lane = threadIdx.x & 31;
  const int wave = threadIdx.x >> 5;
  const int tile = blockIdx.x * (blockDim.x >> 5) + wave;  // 0..4095 (16 px each)

  const int  col = lane & 15;    // D column = output channel; also pixel-in-tile for A
  const bool hi  = lane >= 16;   // upper half-wave: A rows K=2,3

  // Pixel this lane generates A-matrix entries for.
  const int   pix = tile * 16 + col;
  const float px  = (float)(pix & (IMW - 1)) + 0.5f;
  const float py  = (float)(pix >> 8) + 0.5f;

  // Lane-constant 0/1 routing masks for the B matrix.
  // VGPR0 of B holds row K=0 (lanes 0-15) and row K=2 (lanes 16-31);
  // VGPR1 holds row K=1 / K=3.
  const float s0r = (hi ? (col == 7)  : (col == 0)) ? 1.0f : 0.0f;
  const float s0g = (hi ? (col == 8)  : (col == 1)) ? 1.0f : 0.0f;
  const float s0b = (hi ? (col == 9)  : (col == 2)) ? 1.0f : 0.0f;
  const float s0o = (!hi && col == 3) ? 1.0f : 0.0f;        // wsum column of 1s
  const float s1r = (hi ? (col == 10) : (col == 4)) ? 1.0f : 0.0f;
  const float s1g = (hi ? (col == 11) : (col == 5)) ? 1.0f : 0.0f;
  const float s1b = (hi ? (col == 12) : (col == 6)) ? 1.0f : 0.0f;

  v8f acc = {};  // 16x16 f32 accumulator (pixels x channels)

#pragma unroll 4
  for (int n = 0; n < NG; ++n) {
    // Uniform address -> scalar loads through the constant cache.
    const float* p = params + n * 8;
    const float mx = p[0], my = p[1];
    const float ca = p[2], cb = p[3], cc = p[4];
    const float r = p[5], g = p[6], bl = p[7];

    const float dX  = px - mx;
    const float dY  = py - my;
    const float gA  = fmaf(ca, dX, cb * dY);
    const float gB  = fmaf(cb, dX, cc * dY);
    const float sig = 0.5f * fmaf(dX, gA, dY * gB);
    const float w   = __expf(-sig);

    // A matrix: lanes 0-15 -> K rows {0,1} = {w, -gA*w};
    //           lanes 16-31 -> K rows {2,3} = {-gB*w, (gA*gB - b)*w}.
    v2f A;
    A.x = hi ? (-gB * w) : w;
    A.y = hi ? (fmaf(gA, gB, -cb) * w) : (-gA * w);

    // B matrix: per-Gaussian constants routed to the right channel columns.
    v2f B;
    B.x = fmaf(r, s0r, fmaf(g, s0g, fmaf(bl, s0b, s0o)));
    B.y = fmaf(r, s1r, fmaf(g, s1g, bl * s1b));

    // D(16x16,f32) += A(16x4,f32) x B(4x16,f32)
    acc = __builtin_amdgcn_wmma_f32_16x16x4_f32(
        false, A, false, B, (short)0, acc, false, false);
  }

  // D layout: VGPR k holds row M = k (lanes 0-15) / M = k+8 (lanes 16-31),
  // column N = lane&15. Channel plane for channel c starts at c*NPIX.
  if (col < 13) {
    const int base = col * NPIX + tile * 16 + (hi ? 8 : 0);
#pragma unroll
    for (int k = 0; k < 8; ++k) {
      out[base + k] = acc[k];
    }
  }
}

extern "C" void kernel_launch(void* const* d_in, const int* in_sizes, int n_in,
                              void* d_out, int out_size, void* d_ws, size_t ws_size,
                              hipStream_t stream) {
  const float* means      = (const float*)d_in[0];  // (N,2)
  const float* cholesky   = (const float*)d_in[1];  // (N,3)
  const float* rgb_logits = (const float*)d_in[2];  // (N,3)
  float*       out        = (float*)d_out;          // 13 * 65536 floats
  float*       params     = (float*)d_ws;           // 1024 * 8 floats

  gauss2d_precompute<<<(NG + 255) / 256, 256, 0, stream>>>(
      means, cholesky, rgb_logits, params);

  // 4096 tiles of 16 pixels; 8 waves (tiles) per 256-thread block.
  const int tiles  = NPIX / 16;
  const int blocks = tiles / 8;
  gauss2d_splat_wmma<<<blocks, 256, 0, stream>>>(params, out);
}